// ClusteringGroups_14697378087144
// MI455X (gfx1250) — compile-verified
//
#include <hip/hip_runtime.h>

typedef float v2f __attribute__((ext_vector_type(2)));
typedef float v8f __attribute__((ext_vector_type(8)));

#define T_LEN    300
#define T_PAD    320     // 20 tiles of 16 -> every wave owns exactly 5 tiles
#define N_TILES  20
#define TPW      5       // tiles per wave
#define KC       7
#define KPAD     16
#define MAX_IT   100
#define NWL      6
#define BLOCK    128     // 4 wave32 per workgroup

__device__ __constant__ float c_unique_wl[NWL] = {
    3670.69f, 4826.85f, 6223.24f, 7545.98f, 8590.9f, 9710.28f
};

__global__ __launch_bounds__(BLOCK) void
ClusteringGroups_kmeans_kernel(const float* __restrict__ past_values,
                               const float* __restrict__ past_time_features,
                               float* __restrict__ out)
{
    __shared__ float xs[T_PAD];        // masked error series (zero padded)
    __shared__ int   labels[T_PAD];
    __shared__ float cent[KPAD];       // 7 live + 9 zero pad (no select on read)
    __shared__ float sums[KC];
    __shared__ int   cnts[KC];
    __shared__ int   conv;

    const int prob = blockIdx.x;          // 0..6143 = b*6 + w
    const int b    = prob / NWL;
    const int w    = prob % NWL;
    const int tid  = threadIdx.x;
    const int lane = tid & 31;
    const int wave = tid >> 5;
    const bool lo  = (lane < 16);
    const int n    = lane & 15;

    const float wl_target = c_unique_wl[w];

    // Seed L2 with this problem's input rows (gfx1250 global_prefetch_b8 path).
    __builtin_prefetch(&past_time_features[(size_t)b * T_LEN * 2], 0, 1);
    __builtin_prefetch(&past_values[(size_t)b * T_LEN * 2], 0, 1);

    // ---- stage masked series into LDS (b64 loads; .y = stride-2 element) ---
    const float2* __restrict__ pv2  = (const float2*)past_values;
    const float2* __restrict__ ptf2 = (const float2*)past_time_features;
    for (int t = tid; t < T_PAD; t += BLOCK) {
        float v = 0.0f;
        if (t < T_LEN) {
            float wl = ptf2[(size_t)b * T_LEN + t].y;
            float er = pv2[(size_t)b * T_LEN + t].y;
            v = (wl == wl_target) ? er : 0.0f;
        }
        xs[t] = v;
        labels[t] = 0;
    }
    if (tid < KPAD) cent[tid] = 0.0f;     // pad entries stay zero
    __syncthreads();

    // ---- deterministic pseudo-random centroid init ------------------------
    if (tid < KC) {
        unsigned h = (unsigned)prob * 2654435761u ^ ((unsigned)tid * 0x9E3779B9u) ^ 42u;
        h ^= h >> 16; h *= 0x7feb352du;
        h ^= h >> 15; h *= 0x846ca68bu;
        h ^= h >> 16;
        cent[tid] = xs[h % T_LEN];
    }
    __syncthreads();

    // ---- preload this wave's 5 B-operand columns into registers -----------
    // B (4x16 f32), cols = time: B[:,n] = [x[t0+n], 1, 0, 0].  x is constant
    // across all Lloyd iterations, so hold it in VGPRs (pre-masked for the
    // K=2/3 half-wave), keeping the iteration loop free of LDS loads.
    const int tbase = wave * TPW;         // exactly 5 tiles per wave
    float xreg[TPW];
#pragma unroll
    for (int j = 0; j < TPW; ++j)
        xreg[j] = lo ? xs[(tbase + j) * 16 + n] : 0.0f;
    const float bmy = lo ? 1.0f : 0.0f;

    // ---- Lloyd iterations -------------------------------------------------
    for (int it = 0; it < MAX_IT; ++it) {
        // A operand (16x4 f32), rows = clusters: A[m,:] = [-2*c_m, c_m^2, 0, 0].
        float cv = cent[n];               // pad rows read 0 (ignored in D)
        v2f amat;
        amat.x = lo ? (-2.0f * cv) : 0.0f;
        amat.y = lo ? (cv * cv)    : 0.0f;

        // 5 back-to-back WMMAs: D[m][t] = -2*c_m*x[t] + c_m^2
        v8f acc[TPW];
#pragma unroll
        for (int j = 0; j < TPW; ++j) {
            v2f bmat; bmat.x = xreg[j]; bmat.y = bmy;
            v8f z = {};
            acc[j] = __builtin_amdgcn_wmma_f32_16x16x4_f32(
                        /*neg_a=*/false, amat, /*neg_b=*/false, bmat,
                        /*c_mod=*/(short)0, z,
                        /*reuse_a=*/false, /*reuse_b=*/false);
        }

        // Lane L<16 holds score[cluster v][t0+L] in acc[j][v]: in-lane,
        // branch-free argmin over clusters 0..6 (first-min tie-break).
#pragma unroll
        for (int j = 0; j < TPW; ++j) {
            float best = acc[j][0];
            int   ix   = 0;
#pragma unroll
            for (int k = 1; k < KC; ++k) {
                bool tk = acc[j][k] < best;
                best = tk ? acc[j][k] : best;
                ix   = tk ? k         : ix;
            }
            if (lo) labels[(tbase + j) * 16 + n] = ix;  // reconverges after
        }
        __syncthreads();

        if (tid < KC) { sums[tid] = 0.0f; cnts[tid] = 0; }
        if (tid == 0) conv = 1;
        __syncthreads();

        for (int t = tid; t < T_LEN; t += BLOCK) {
            int l = labels[t];
            atomicAdd(&sums[l], xs[t]);     // ds_add_f32
            atomicAdd(&cnts[l], 1);         // ds_add_u32
        }
        __syncthreads();

        if (tid < KC) {
            float oc = cent[tid];
            float nc = (cnts[tid] > 0) ? (sums[tid] / (float)cnts[tid]) : oc;
            if (nc != oc) atomicAnd(&conv, 0);
            cent[tid] = nc;
        }
        __syncthreads();

        if (conv) break;
        __syncthreads();
    }

    // ---- emit one-hot (K, T): ds_load_b128 in, b128 stores out ------------
    // 2100 floats per problem; 8400-byte stride keeps 16B alignment.
    float4* __restrict__ out4 = (float4*)(out + (size_t)prob * (KC * T_LEN));
    for (int i = tid; i < (KC * T_LEN) / 4; i += BLOCK) {   // 525 float4s
        const int e0 = i * 4;
        const int k  = e0 / T_LEN;
        const int t  = e0 - k * T_LEN;     // T_LEN % 4 == 0: no row crossing
        int4 l4 = *(const int4*)&labels[t];
        float4 r;
        r.x = (l4.x == k) ? 1.0f : 0.0f;
        r.y = (l4.y == k) ? 1.0f : 0.0f;
        r.z = (l4.z == k) ? 1.0f : 0.0f;
        r.w = (l4.w == k) ? 1.0f : 0.0f;
        out4[i] = r;
    }
}

extern "C" void kernel_launch(void* const* d_in, const int* in_sizes, int n_in,
                              void* d_out, int out_size, void* d_ws, size_t ws_size,
                              hipStream_t stream) {
    const float* past_values        = (const float*)d_in[0];
    const float* past_time_features = (const float*)d_in[1];
    float* out = (float*)d_out;

    const int n_problems = 1024 * NWL;   // 6144 independent k-means problems
    dim3 grid(n_problems), block(BLOCK);
    hipLaunchKernelGGL(ClusteringGroups_kmeans_kernel, grid, block, 0, stream,
                       past_values, past_time_features, out);
}